// TSBRNN_42631845380194
// MI455X (gfx1250) — compile-verified
//
#include <hip/hip_runtime.h>
#include <cstdint>

// ---------------------------------------------------------------------------
// TSB RNN scan: B=8192 rows, T=4096 steps. Bandwidth-bound segmented affine
// scan in 3 phases. Tile stage-in via Tensor Data Mover (tensor_load_to_lds,
// with hardware LDS row padding); stage-out via async-from-LDS stores.
// ---------------------------------------------------------------------------

namespace {
constexpr int kB      = 8192;
constexpr int kT      = 4096;
constexpr int kL      = 64;            // scan chunk length (even -> A/B decouple)
constexpr int kChunks = kT / kL;       // 64
constexpr int kTileR  = 64;            // rows per block tile
constexpr int kTileT  = 128;           // timesteps per block tile (2 chunks)
constexpr int kStride = 132;           // LDS row stride (floats); TDM pad 4 DW/row
constexpr int kThr    = 128;           // kTileR * (kTileT/kL) threads = 4 waves
}

typedef __attribute__((ext_vector_type(4))) uint32_t uint4v;
typedef __attribute__((ext_vector_type(8))) uint32_t uint8v;

__device__ __forceinline__ uint32_t lds_off_of(const void* p) {
  // Generic pointer into LDS: low 32 bits are the LDS byte address
  // (flat LDS aperture truncates upper bits).
  return (uint32_t)(uintptr_t)p;
}

__device__ __forceinline__ void async_load_b128(uint32_t lds, const float* g) {
  asm volatile("global_load_async_to_lds_b128 %0, %1, off"
               :: "v"(lds), "v"((uint64_t)(uintptr_t)g)
               : "memory");
}

__device__ __forceinline__ void async_store_b128(float* g, uint32_t lds) {
  asm volatile("global_store_async_from_lds_b128 %0, %1, off"
               :: "v"((uint64_t)(uintptr_t)g), "v"(lds)
               : "memory");
}

__device__ __forceinline__ void wait_asynccnt0() {
  asm volatile("s_wait_asynccnt 0" ::: "memory");
}

// ---------------------------------------------------------------------------
// TDM: load a 2D tile (kTileR rows x kTileT f32) whose global row stride is
// kT elements into LDS at lds_base, padding each 128-DWORD row by 4 DWORDs
// (-> LDS stride 132 floats, bank-conflict-friendly and 16B aligned).
// D# per CDNA5 ISA (group0 + group1, 2D form, VADDR2/3=NULL).
// ---------------------------------------------------------------------------
__device__ __forceinline__ void tdm_load_tile(uint32_t lds_base, const float* gtile) {
  const uint64_t ga = (uint64_t)(uintptr_t)gtile;

  uint4v g0;
  g0.x = 1u;                                     // count=1 (valid), user mode
  g0.y = lds_base;                               // lds_addr (bytes)
  g0.z = (uint32_t)ga;                           // global_addr[31:0] (tile start)
  g0.w = (uint32_t)((ga >> 32) & 0x01FFFFFFu)    // global_addr[56:32]
       | (2u << 30);                             // type=2 ("image")

  uint8v g1;
  g1.s0 = (2u << 16)                             // data_size: 4 bytes
        | (1u << 20)                             // pad_enable
        | (6u << 22)                             // pad_interval: every 128 DWORDs
        | (3u << 25);                            // pad_amount: 4 DWORDs
  g1.s1 = (uint32_t)kT << 16;                    // tensor_dim0[15:0] = 4096
  g1.s2 = (uint32_t)kB << 16;                    // tensor_dim1[15:0] = 8192
  g1.s3 = (uint32_t)kTileT << 16;                // tile_dim0 = 128 elements
  g1.s4 = (uint32_t)kTileR;                      // tile_dim1 = 64 rows (tile_dim2=0)
  g1.s5 = (uint32_t)kT;                          // tensor_dim0_stride = 4096 elems
  g1.s6 = 0u;                                    // stride0 hi, dim1_stride lo
  g1.s7 = 0u;                                    // dim1_stride hi

  asm volatile("tensor_load_to_lds %0, %1"
               :: "s"(g0), "s"(g1)
               : "memory");
}

// Cooperative tile stage-in: TDM moves the tile; one async b128 per lane of
// wave 0 fetches the row-0 slice (X_t) into xrow.
__device__ __forceinline__ void stage_tile(const float* __restrict__ x,
                                           float* tile, float* xrow,
                                           int r0, int t0, int lane, int wv) {
  if (wv == 0) {
    tdm_load_tile(lds_off_of(tile), x + (size_t)r0 * kT + t0);
    async_load_b128(lds_off_of(&xrow[lane * 4]), x + t0 + lane * 4);
    wait_asynccnt0();
    __builtin_amdgcn_s_wait_tensorcnt(0);
  }
  __syncthreads();
}

// ---------------------------------------------------------------------------
// Phase 1: per (row, chunk) compose the chunk's affine maps.
// A-chain: A_end = pa*A_start + qa ;  B-chain: B_end = pb*B_start + qb
// ---------------------------------------------------------------------------
__global__ __launch_bounds__(kThr)
void tsb_phase1(const float* __restrict__ x, const float* __restrict__ alphap,
                const float* __restrict__ betap, float4* __restrict__ ws1) {
  __shared__ __align__(16) float tile[kTileR * kStride];
  __shared__ __align__(16) float xrow[kTileT];

  const int tid  = threadIdx.x;
  const int lane = tid & 31;
  const int wv   = tid >> 5;
  const int r0   = blockIdx.y * kTileR;
  const int t0   = blockIdx.x * kTileT;

  stage_tile(x, tile, xrow, r0, t0, lane, wv);

  const float a   = alphap[0];
  const float bt  = betap[0];
  const float oma = 1.0f - a;
  const float omb = 1.0f - bt;

  const int k    = tid >> 6;           // chunk within tile (0..1)
  const int r    = tid & 63;           // row within tile
  const int base = r * kStride + k * kL;
  const int xb   = k * kL;

  float pa = 1.0f, qa = 0.0f, pb = 1.0f, qb = 0.0f;
#pragma unroll 4
  for (int i = 0; i < kL; ++i) {
    const float xv = tile[base + i];
    const float Xv = xrow[xb + i];
    const bool  nz = (xv != 0.0f);
    const float c  = nz ? oma : 1.0f;        // B' = c*A + d
    const float d  = nz ? a * Xv : 0.0f;
    const float npa = omb * pb;              // A' = (1-b)*B + b*nz (maps swap)
    const float nqa = omb * qb + (nz ? bt : 0.0f);
    const float npb = c * pa;
    const float nqb = c * qa + d;
    pa = npa; qa = nqa; pb = npb; qb = nqb;
  }

  const int cidx = blockIdx.x * (kTileT / kL) + k;
  const int bidx = r0 + r;
  ws1[(size_t)cidx * kB + bidx] = make_float4(pa, qa, pb, qb);  // coalesced b128
}

// ---------------------------------------------------------------------------
// Phase 2: per row, sequentially scan the 64 chunk maps; emit per-chunk start
// states (A,B). Fully coalesced float4 reads / float2 writes along rows.
// ---------------------------------------------------------------------------
__global__ __launch_bounds__(256)
void tsb_phase2(const float4* __restrict__ ws1, float2* __restrict__ ws2) {
  const int b = blockIdx.x * 256 + threadIdx.x;
  float A = 0.0f, Bv = 0.0f;                  // Z0 = P0 = 0
  for (int c = 0; c < kChunks; ++c) {
    ws2[(size_t)c * kB + b] = make_float2(A, Bv);
    const float4 m = ws1[(size_t)c * kB + b];
    const float An = m.x * A + m.y;
    const float Bn = m.z * Bv + m.w;
    A = An; Bv = Bn;
  }
}

// ---------------------------------------------------------------------------
// Phase 3: replay each chunk from its start state, overwrite LDS tile with
// out = A'*B', async-store the tile back (coalesced).
// ---------------------------------------------------------------------------
__global__ __launch_bounds__(kThr)
void tsb_phase3(const float* __restrict__ x, const float* __restrict__ alphap,
                const float* __restrict__ betap, const float2* __restrict__ ws2,
                float* __restrict__ out) {
  __shared__ __align__(16) float tile[kTileR * kStride];
  __shared__ __align__(16) float xrow[kTileT];

  const int tid  = threadIdx.x;
  const int lane = tid & 31;
  const int wv   = tid >> 5;
  const int r0   = blockIdx.y * kTileR;
  const int t0   = blockIdx.x * kTileT;

  stage_tile(x, tile, xrow, r0, t0, lane, wv);

  const float a   = alphap[0];
  const float bt  = betap[0];
  const float oma = 1.0f - a;
  const float omb = 1.0f - bt;

  const int k    = tid >> 6;
  const int r    = tid & 63;
  const int base = r * kStride + k * kL;
  const int xb   = k * kL;

  const int cidx = blockIdx.x * (kTileT / kL) + k;
  const int bidx = r0 + r;
  const float2 st = ws2[(size_t)cidx * kB + bidx];
  float A = st.x, Bv = st.y;

#pragma unroll 4
  for (int i = 0; i < kL; ++i) {
    const float xv = tile[base + i];
    const float Xv = xrow[xb + i];
    const bool  nz = (xv != 0.0f);
    const float An = omb * Bv + (nz ? bt : 0.0f);       // P_next
    const float Bn = nz ? (a * Xv + oma * A) : A;       // Z_next (X = row 0!)
    tile[base + i] = An * Bn;                           // out = P_next * Z_next
    A = An; Bv = Bn;
  }
  __syncthreads();

  for (int rr = wv; rr < kTileR; rr += 4) {
    float* g = out + (size_t)(r0 + rr) * kT + t0 + lane * 4;
    async_store_b128(g, lds_off_of(&tile[rr * kStride + lane * 4]));
  }
  wait_asynccnt0();
}

// ---------------------------------------------------------------------------
extern "C" void kernel_launch(void* const* d_in, const int* in_sizes, int n_in,
                              void* d_out, int out_size, void* d_ws, size_t ws_size,
                              hipStream_t stream) {
  (void)in_sizes; (void)n_in; (void)out_size; (void)ws_size;

  const float* x     = (const float*)d_in[0];
  const float* alpha = (const float*)d_in[1];
  const float* beta  = (const float*)d_in[2];
  float*       out   = (float*)d_out;

  float4* ws1 = (float4*)d_ws;                                         // 8 MiB
  float2* ws2 = (float2*)((char*)d_ws +
                          (size_t)kChunks * kB * sizeof(float4));      // +4 MiB

  dim3 gridA(kT / kTileT, kB / kTileR);   // 32 x 128 = 4096 blocks
  tsb_phase1<<<gridA, kThr, 0, stream>>>(x, alpha, beta, ws1);
  tsb_phase2<<<kB / 256, 256, 0, stream>>>(ws1, ws2);
  tsb_phase3<<<gridA, kThr, 0, stream>>>(x, alpha, beta, ws2, out);
}